// OptimizedLinear_79912161509906
// MI455X (gfx1250) — compile-verified
//
#include <hip/hip_runtime.h>
#include <hip/hip_bf16.h>
#include <math.h>

// CDNA5 (gfx1250) wave32 WMMA types
typedef float v2f __attribute__((ext_vector_type(2)));
typedef float v8f __attribute__((ext_vector_type(8)));

// Irrep linear: y[n, off + o*D + c] = alpha * sum_i x[n, off + i*D + c] * W[i, o]
//
// One wave -> 16 samples (M tile) x full MUL output columns, one component c.
// K-loop in steps of 4 using V_WMMA_F32_16X16X4_F32 (native fp32 matrix path).
//
// W is staged in LDS once per workgroup, pair-interleaved so a B fragment's
// two K-rows (k, k+1) at the same column are adjacent -> one ds_load_b64.
// STRIDE2 = 2*MUL+32 dwords: lane-halves land 32 banks apart => conflict-free.
//
// Fragment layouts per CDNA5 ISA 7.12.2 (wave32):
//   A (16x4):  lanes 0-15 hold M=lane; VGPR0 = K {0 | 2}, VGPR1 = K {1 | 3}
//   B (4x16):  lanes 0-15 hold N=lane; VGPR0 = K {0 | 2}, VGPR1 = K {1 | 3}
//   C/D v8f:   VGPR r: lanes 0-15 -> M=r, lanes 16-31 -> M=r+8; N = lane&15
template <int MUL, int D>
__global__ __launch_bounds__(256)
void irrep_linear_wmma(const float* __restrict__ x,
                       const float* __restrict__ W,
                       float* __restrict__ y,
                       int N, int off, float alpha)
{
    constexpr int DIM     = 480;          // total irreps dim (row stride)
    constexpr int OT      = MUL / 16;     // 16-wide output tiles per wave
    constexpr int KSTEPS  = MUL / 4;      // k-steps of 4
    constexpr int PAIRS   = MUL / 2;      // interleaved K-row pairs
    constexpr int STRIDE2 = 2 * MUL + 32; // dwords per pair-row (bank padding)

    __shared__ float lw[PAIRS * STRIDE2];

    // ---- Stage W into LDS (pair-interleaved), all 256 threads, coalesced ----
    for (int i = threadIdx.x; i < MUL * MUL; i += 256) {
        const int k   = i / MUL;
        const int col = i - k * MUL;
        lw[(k >> 1) * STRIDE2 + col * 2 + (k & 1)] = W[i];
    }
    __syncthreads();

    const int lane   = threadIdx.x & 31;
    const int waveId = blockIdx.x * (blockDim.x >> 5) + (threadIdx.x >> 5);

    const int numMTiles = (N + 15) >> 4;
    const int mTile = waveId / D;          // wave-uniform
    const int c     = waveId % D;          // component index (wave-uniform)

    if (mTile < numMTiles) {               // uniform guard: EXEC all-1s inside
        const int halfk = lane >> 4;       // which K pair of the quad this lane holds
        const int lid   = lane & 15;

        int row = mTile * 16 + lid;
        if (row >= N) row = N - 1;         // clamp loads; stores guarded later

        const float* __restrict__ abase = x + (size_t)row * DIM + off + c;

        v8f acc[OT] = {};

        #pragma unroll 4
        for (int ks = 0; ks < KSTEPS; ++ks) {
            const int k = ks * 4 + 2 * halfk;      // this lane-half's K pair {k, k+1}
            const int p = ks * 2 + halfk;          // interleaved pair index (k>>1)
            v2f a;
            a.x = abase[(size_t)k * D];
            a.y = abase[(size_t)(k + 1) * D];
            const float* __restrict__ brow = &lw[p * STRIDE2 + lid * 2];
            #pragma unroll
            for (int t = 0; t < OT; ++t) {
                // (W[k, col], W[k+1, col]) adjacent in LDS -> ds_load_b64
                v2f b = *(const v2f*)(brow + t * 32);
                // 8 args: (neg_a, A, neg_b, B, c_mod, C, reuse_a, reuse_b)
                acc[t] = __builtin_amdgcn_wmma_f32_16x16x4_f32(
                    false, a, false, b, (short)0, acc[t], false, false);
            }
        }

        // Write back: acc[t][r] holds (m = r + 8*halfk, o = t*16 + lid)
        #pragma unroll
        for (int t = 0; t < OT; ++t) {
            const int o = t * 16 + lid;
            float* __restrict__ ybase = y + off + (size_t)o * D + c;
            #pragma unroll
            for (int r = 0; r < 8; ++r) {
                const int n = mTile * 16 + r + 8 * halfk;
                if (n < N)
                    ybase[(size_t)n * DIM] = acc[t][r] * alpha;
            }
        }
    }
}

extern "C" void kernel_launch(void* const* d_in, const int* in_sizes, int n_in,
                              void* d_out, int out_size, void* d_ws, size_t ws_size,
                              hipStream_t stream) {
    const float* x  = (const float*)d_in[0];  // [N, 480]
    const float* W0 = (const float*)d_in[1];  // [128, 128]
    const float* W1 = (const float*)d_in[2];  // [64, 64]
    const float* W2 = (const float*)d_in[3];  // [32, 32]
    float* y = (float*)d_out;                 // [N, 480]

    const int N = in_sizes[0] / 480;
    const int numMTiles = (N + 15) / 16;

    // 8 waves per 256-thread block; one wave per (mTile, component)
    auto blocksFor = [&](int d) { return (numMTiles * d + 7) / 8; };

    const float a0 = 1.0f / sqrtf(128.0f);
    const float a1 = 1.0f / sqrtf(64.0f);
    const float a2 = 1.0f / sqrtf(32.0f);

    irrep_linear_wmma<128, 1><<<blocksFor(1), 256, 0, stream>>>(x, W0, y, N,   0, a0);
    irrep_linear_wmma< 64, 3><<<blocksFor(3), 256, 0, stream>>>(x, W1, y, N, 128, a1);
    irrep_linear_wmma< 32, 5><<<blocksFor(5), 256, 0, stream>>>(x, W2, y, N, 320, a2);
}